// ContrastiveLoss_29394756173978
// MI455X (gfx1250) — compile-verified
//
#include <hip/hip_runtime.h>
#include <hip/hip_bf16.h>
#include <math.h>

// CDNA5 / gfx1250 fused supervised-contrastive loss.
//   sim = P P^T / T,  per-row masked online logsumexp + positive-pair sums,
//   loss = sum_pos(lse_i - sim_ij) / num_pos
// fp32 GEMM via V_WMMA_F32_16X16X4_F32 (matches fp32 reference numerics).

typedef __attribute__((ext_vector_type(2))) float v2f;
typedef __attribute__((ext_vector_type(8))) float v8f;

#define D_DIM       256
#define KSTEPS      64           // 256 / 4
#define LDS_STRIDE  260          // 260 % 64 == 4 -> conflict-free ds_load_b64
#define ROWS_PER_WAVE 16
#define WAVES_PER_BLOCK 2
#define ROWS_PER_BLOCK (ROWS_PER_WAVE * WAVES_PER_BLOCK)
#define INV_T 14.285714285714286f   // 1 / 0.07

__global__ void cl_init_ws(float* ws) {
    ws[0] = 0.0f;   // total loss numerator
    ws[1] = 0.0f;   // positive pair count
}

__global__ __launch_bounds__(WAVES_PER_BLOCK * 32)
void ContrastiveLoss_29394756173978_kernel(const float* __restrict__ P,
                                           const int*   __restrict__ aff,
                                           const int*   __restrict__ inst,
                                           float*       __restrict__ ws,
                                           int B) {
    __shared__ float As[ROWS_PER_BLOCK * LDS_STRIDE];  // 32 x 256 (padded)
    __shared__ float Bs[16 * LDS_STRIDE];              // 16 x 256 (padded)

    const int tid  = threadIdx.x;
    const int lane = tid & 31;
    const int wave = tid >> 5;
    const int half = lane >> 4;        // 0: lanes 0-15, 1: lanes 16-31
    const int nidx = lane & 15;        // M-index for A-frag, N-index for B/C frags

    const int rowBase     = blockIdx.x * ROWS_PER_BLOCK;
    const int waveRowBase = rowBase + wave * ROWS_PER_WAVE;

    // ---- Stage this block's 32 A rows into LDS (float4 copies) ----
    for (int f = tid; f < ROWS_PER_BLOCK * (D_DIM / 4); f += WAVES_PER_BLOCK * 32) {
        const int r  = f >> 6;          // D_DIM/4 == 64 float4 per row
        const int c4 = f & 63;
        const float4 v = ((const float4*)P)[(size_t)(rowBase + r) * (D_DIM / 4) + c4];
        *(float4*)&As[r * LDS_STRIDE + c4 * 4] = v;
    }

    // ---- Per-lane row metadata: 8 rows per lane (C-layout: row = g + 8*half) ----
    int rAff[8], rInst[8];
#pragma unroll
    for (int g = 0; g < 8; ++g) {
        const int row = waveRowBase + g + 8 * half;
        rAff[g]  = aff[row];
        rInst[g] = inst[row];
    }

    // Online logsumexp + positive accumulators (per lane: 8 rows, columns == nidx mod 16)
    float m_[8], s_[8], ps_[8], pc_[8];
#pragma unroll
    for (int g = 0; g < 8; ++g) { m_[g] = -INFINITY; s_[g] = 0.f; ps_[g] = 0.f; pc_[g] = 0.f; }

    __syncthreads();

    const float* aBase = &As[(wave * ROWS_PER_WAVE + nidx) * LDS_STRIDE];
    const float* bBase = &Bs[nidx * LDS_STRIDE];

    // ---- Sweep all column tiles ----
    for (int j0 = 0; j0 < B; j0 += 16) {
        // Stage B tile (16 rows of P) cooperatively
        for (int f = tid; f < 16 * (D_DIM / 4); f += WAVES_PER_BLOCK * 32) {
            const int r  = f >> 6;
            const int c4 = f & 63;
            const float4 v = ((const float4*)P)[(size_t)(j0 + r) * (D_DIM / 4) + c4];
            *(float4*)&Bs[r * LDS_STRIDE + c4 * 4] = v;
        }
        __syncthreads();

        // 16x16 fp32 tile of P P^T via 64 chained V_WMMA_F32_16X16X4_F32
        v8f acc = {};
#pragma unroll
        for (int kk = 0; kk < KSTEPS; ++kk) {
            const int k = kk * 4 + 2 * half;
            v2f a = *(const v2f*)&aBase[k];   // A[m=nidx][k..k+1] of this wave's strip
            v2f b = *(const v2f*)&bBase[k];   // P[j0+nidx][k..k+1] == B[k..k+1][n=nidx]
            acc = __builtin_amdgcn_wmma_f32_16x16x4_f32(
                false, a, false, b, (short)0, acc, false, false);
        }
        __syncthreads();   // Bs free for restage; epilogue below is register-only

        // Fused epilogue: per lane handles column (j0+nidx) for its 8 rows
        const int col   = j0 + nidx;
        const int cAff  = aff[col];
        const int cInst = inst[col];
#pragma unroll
        for (int g = 0; g < 8; ++g) {
            const int row = waveRowBase + g + 8 * half;
            const float x = acc[g] * INV_T;
            if (row != col) {                    // self-mask -> -inf (skip)
                const float mo = m_[g];
                const float mn = fmaxf(mo, x);
                s_[g] = s_[g] * __expf(mo - mn) + __expf(x - mn);
                m_[g] = mn;
                if ((rAff[g] == cAff) && (rInst[g] != cInst)) {   // positive pair
                    ps_[g] += x;
                    pc_[g] += 1.0f;
                }
            }
        }
    }

    // ---- Combine across the 16 lanes of each half (wave32 xor shuffles) ----
#pragma unroll
    for (int mask = 1; mask <= 8; mask <<= 1) {
#pragma unroll
        for (int g = 0; g < 8; ++g) {
            const float mo = m_[g], so = s_[g];
            const float m2 = __shfl_xor(mo, mask, 32);
            const float s2 = __shfl_xor(so, mask, 32);
            const float mn = fmaxf(mo, m2);
            s_[g] = so * __expf(mo - mn) + s2 * __expf(m2 - mn);
            m_[g] = mn;
            ps_[g] += __shfl_xor(ps_[g], mask, 32);
            pc_[g] += __shfl_xor(pc_[g], mask, 32);
        }
    }

    // Lane 0 holds rows base..base+7, lane 16 holds rows base+8..base+15
    if (nidx == 0) {
        float tot = 0.f, cnt = 0.f;
#pragma unroll
        for (int g = 0; g < 8; ++g) {
            const float lse = m_[g] + __logf(s_[g]);
            tot += pc_[g] * lse - ps_[g];
            cnt += pc_[g];
        }
        atomicAdd(&ws[0], tot);
        atomicAdd(&ws[1], cnt);
    }
}

__global__ void cl_finalize(const float* ws, float* out) {
    const float cnt = ws[1];
    out[0] = (cnt > 0.f) ? (ws[0] / cnt) : 0.0f;
}

extern "C" void kernel_launch(void* const* d_in, const int* in_sizes, int n_in,
                              void* d_out, int out_size, void* d_ws, size_t ws_size,
                              hipStream_t stream) {
    const float* P    = (const float*)d_in[0];
    const int*   aff  = (const int*)d_in[1];
    const int*   inst = (const int*)d_in[2];
    float*       ws   = (float*)d_ws;
    float*       out  = (float*)d_out;

    const int B = in_sizes[1];                 // 8192
    const int grid = B / ROWS_PER_BLOCK;       // 256 blocks

    cl_init_ws<<<1, 1, 0, stream>>>(ws);
    ContrastiveLoss_29394756173978_kernel<<<grid, WAVES_PER_BLOCK * 32, 0, stream>>>(
        P, aff, inst, ws, B);
    cl_finalize<<<1, 1, 0, stream>>>(ws, out);
}